// TransformerBlock_44865228374187
// MI455X (gfx1250) — compile-verified
//
#include <hip/hip_runtime.h>
#include <math.h>

// ---- problem constants ----
#define BATCH 32
#define DMODEL 2048
#define NHEADS 16
#define NKV 2
#define HDIM 128
#define SEQL 4096
#define IDIM 11008
#define EPS 1e-6f
#define KC 256  // K-chunk staged in LDS per block (KC*32 floats = 32 KB)

typedef __attribute__((ext_vector_type(2))) float v2f;
typedef __attribute__((ext_vector_type(8))) float v8f;
typedef __attribute__((ext_vector_type(4))) unsigned u32x4;
typedef __attribute__((ext_vector_type(8))) unsigned u32x8;

// Activation panels are stored "pair-interleaved K-major":
//   f(k,m) = (k>>1)*(2*BATCH) + 2*m + (k&1)
// so a lane's two consecutive-K A-tile values are adjacent (one b64 load).
__device__ __forceinline__ size_t apack(int k, int m) {
  return (size_t)(k >> 1) * (2 * BATCH) + 2 * m + (k & 1);
}

// ---------------------------------------------------------------------------
// TDM: DMA one contiguous 32 KB activation chunk (KC*32 fp32) into LDS.
// D# built per CDNA5 ISA 8.3/8.4: flat 1-D tile (tile_dim0 = 8192, dim1 = 1),
// data_size = 4B, no multicast / pad / iterate. Issued by one wave per block;
// completion tracked with TENSORcnt.
// ---------------------------------------------------------------------------
__device__ __forceinline__ void tdm_load_chunk(const float* gsrc, unsigned lds_off) {
  const unsigned nElem = KC * BATCH;  // 8192 elements
  unsigned long long ga = (unsigned long long)(size_t)gsrc;
  u32x4 g0;
  g0[0] = 1u;                                              // count=1, user mode
  g0[1] = lds_off;                                         // lds_addr (bytes)
  g0[2] = (unsigned)(ga & 0xFFFFFFFFu);                    // global_addr[31:0]
  g0[3] = (unsigned)((ga >> 32) & 0x1FFFFFFu) | (2u << 30);// addr[56:32] | type=2
  u32x8 g1;
  g1[0] = 2u << 16;                    // workgroup_mask=0, data_size=2 (4 bytes)
  g1[1] = (nElem & 0xFFFFu) << 16;     // tensor_dim0[15:0]
  g1[2] = ((nElem >> 16) & 0xFFFFu) | (1u << 16);  // tensor_dim0[31:16], tensor_dim1=1
  g1[3] = (nElem & 0xFFFFu) << 16;     // tensor_dim1[31:16]=0, tile_dim0 = nElem
  g1[4] = 1u;                          // tile_dim1 = 1, tile_dim2 = 0
  g1[5] = nElem;                       // tensor_dim0_stride[31:0]
  g1[6] = 0u;                          // stride0[47:32]=0, stride1[15:0]=0
  g1[7] = 0u;                          // stride1[47:16]=0
  asm volatile("tensor_load_to_lds %0, %1" :: "s"(g0), "s"(g1) : "memory");
}

// ---------------------------------------------------------------------------
// RMSNorm: reads x[32][2048], writes pair-interleaved transposed panel
// ---------------------------------------------------------------------------
__global__ __launch_bounds__(256) void rmsnorm_T_kernel(
    const float* __restrict__ x, const float* __restrict__ w,
    float* __restrict__ outT) {
  __shared__ float red[9];
  const int b = blockIdx.x;
  const float* xb = x + (size_t)b * DMODEL;
  float s = 0.f;
  for (int i = threadIdx.x; i < DMODEL; i += 256) { float v = xb[i]; s += v * v; }
  for (int off = 16; off > 0; off >>= 1) s += __shfl_xor(s, off, 32);
  if ((threadIdx.x & 31) == 0) red[threadIdx.x >> 5] = s;
  __syncthreads();
  if (threadIdx.x == 0) {
    float t = 0.f;
    for (int i = 0; i < 8; ++i) t += red[i];
    red[8] = rsqrtf(t / (float)DMODEL + EPS);
  }
  __syncthreads();
  const float scale = red[8];
  for (int i = threadIdx.x; i < DMODEL; i += 256)
    outT[apack(i, b)] = xb[i] * scale * w[i];
}

// ---------------------------------------------------------------------------
// fp32 WMMA GEMM with TDM-staged, double-buffered activation panel:
//   out[32][N] = A[32][K] * W[N][K]^T (+bias)(+resid)
// A panel in pair-interleaved layout (see apack). One wave = 32x16 tile
// (two V_WMMA_F32_16X16X4_F32 accumulators); 8 waves/block share the LDS panel.
// Wave 0 DMAs chunk c+1 while everyone computes chunk c from LDS.
// ---------------------------------------------------------------------------
__global__ __launch_bounds__(256) void gemm_wmma_kernel(
    const float* __restrict__ AT,    // pair-interleaved [K][32]
    const float* __restrict__ W,     // [N][K] row-major
    const float* __restrict__ bias,  // [N] or nullptr
    const float* __restrict__ resid, // [32][N] or nullptr
    float* __restrict__ out,         // [32][N]
    int K, int N) {
  __shared__ float abuf[2][KC * BATCH];  // 2 x 32 KB
  const int wv = threadIdx.x >> 5;
  const int lane = threadIdx.x & 31;
  const int l16 = lane & 15;
  const int half = lane >> 4;
  const int koff = half * 2;  // lanes 0-15 hold K={k,k+1}; 16-31 K={k+2,k+3}
  const int n0 = (blockIdx.x * 8 + wv) * 16;
  const int nrow = (n0 + l16 < N) ? (n0 + l16) : (N - 1);
  const float* wrow = W + (size_t)nrow * K + koff;

  const unsigned lds0 = (unsigned)(size_t)&abuf[0][0];
  const unsigned lds1 = (unsigned)(size_t)&abuf[1][0];
  const int nchunks = K / KC;

  if (wv == 0) tdm_load_chunk(AT, lds0);

  v8f c0 = {}; v8f c1 = {};
  for (int c = 0; c < nchunks; ++c) {
    if (wv == 0) {
      if (c + 1 < nchunks) {
        tdm_load_chunk(AT + (size_t)(c + 1) * (KC * BATCH),
                       ((c + 1) & 1) ? lds1 : lds0);
        __builtin_amdgcn_s_wait_tensorcnt(1);  // chunk c complete (in-order)
      } else {
        __builtin_amdgcn_s_wait_tensorcnt(0);
      }
    }
    __syncthreads();  // LDS panel for chunk c visible to all waves
    const float* al = abuf[c & 1];
    const float* wptr = wrow + (size_t)c * KC;
#pragma unroll 4
    for (int kl = 0; kl < KC; kl += 4) {
      const int kp = (kl + koff) >> 1;  // K-pair index for this lane
      v2f a0 = *(const v2f*)(al + kp * (2 * BATCH) + 2 * l16);        // M-tile 0
      v2f a1 = *(const v2f*)(al + kp * (2 * BATCH) + 32 + 2 * l16);   // M-tile 1
      v2f bb = *(const v2f*)(wptr + kl);
      c0 = __builtin_amdgcn_wmma_f32_16x16x4_f32(false, a0, false, bb, (short)0, c0, false, false);
      c1 = __builtin_amdgcn_wmma_f32_16x16x4_f32(false, a1, false, bb, (short)0, c1, false, false);
    }
    __syncthreads();  // everyone done with buf[c&1] before it is re-filled
  }

  const int n = n0 + l16;
  if (n < N) {
    const float bsv = bias ? bias[n] : 0.f;
#pragma unroll
    for (int r = 0; r < 8; ++r) {
      const int m0 = half * 8 + r;  // C/D layout rows, M-tile 0
      const int m1 = 16 + m0;       // M-tile 1
      float v0 = c0[r] + bsv;
      float v1 = c1[r] + bsv;
      if (resid) {
        v0 += resid[(size_t)m0 * N + n];
        v1 += resid[(size_t)m1 * N + n];
      }
      out[(size_t)m0 * N + n] = v0;
      out[(size_t)m1 * N + n] = v1;
    }
  }
}

// ---------------------------------------------------------------------------
// RoPE (position L-1) applied in place to q[32][16*128] and k[32][2*128]
// ---------------------------------------------------------------------------
__global__ __launch_bounds__(256) void rope_kernel(
    float* __restrict__ q, float* __restrict__ kb, const float* __restrict__ fc) {
  const int QP = BATCH * NHEADS * (HDIM / 2);
  const int KP = BATCH * NKV * (HDIM / 2);
  int idx = blockIdx.x * blockDim.x + threadIdx.x;
  if (idx >= QP + KP) return;
  float* t;
  int pair;
  if (idx < QP) {
    int b = idx / (NHEADS * (HDIM / 2));
    int rem = idx % (NHEADS * (HDIM / 2));
    int h = rem / (HDIM / 2);
    pair = rem % (HDIM / 2);
    t = q + (size_t)b * (NHEADS * HDIM) + h * HDIM + 2 * pair;
  } else {
    int j = idx - QP;
    int b = j / (NKV * (HDIM / 2));
    int rem = j % (NKV * (HDIM / 2));
    int h = rem / (HDIM / 2);
    pair = rem % (HDIM / 2);
    t = kb + (size_t)b * (NKV * HDIM) + h * HDIM + 2 * pair;
  }
  const float c = fc[2 * pair], s = fc[2 * pair + 1];
  const float xr = t[0], xi = t[1];
  t[0] = xr * c - xi * s;
  t[1] = xr * s + xi * c;
}

// ---------------------------------------------------------------------------
// new_cache = concat(cache[:,1:], new_kv)  — float4 streaming copy
// ---------------------------------------------------------------------------
__global__ __launch_bounds__(256) void cache_shift_kernel(
    const float4* __restrict__ ck, const float4* __restrict__ cv,
    const float4* __restrict__ kb, const float4* __restrict__ vb,
    float4* __restrict__ nk, float4* __restrict__ nv) {
  const size_t per = (size_t)BATCH * SEQL * NKV * HDIM / 4;
  size_t idx = (size_t)blockIdx.x * blockDim.x + threadIdx.x;
  const bool isV = idx >= per;
  size_t g = isV ? idx - per : idx;
  if (g >= per) return;
  const int row4 = NKV * HDIM / 4;  // 64
  int b = (int)(g / ((size_t)SEQL * row4));
  int rem = (int)(g % ((size_t)SEQL * row4));
  int pos = rem / row4;
  int c = rem % row4;
  const float4* cache = isV ? cv : ck;
  const float4* last = isV ? vb : kb;
  float4 val;
  if (pos < SEQL - 1)
    val = cache[(size_t)b * SEQL * row4 + (size_t)(pos + 1) * row4 + c];
  else
    val = last[b * row4 + c];
  (isV ? nv : nk)[g] = val;
}

// ---------------------------------------------------------------------------
// Decode attention: block per (b, kv-head), all 8 GQA query heads together so
// each K/V cache row streams once per group. Reads the original cache with a
// +1 position shift; position L-1 comes from fresh rope'd k/v. Writes the
// pair-interleaved transposed panel for the wo GEMM.
// ---------------------------------------------------------------------------
__global__ __launch_bounds__(256) void attn_kernel(
    const float* __restrict__ q,
    const float* __restrict__ ck, const float* __restrict__ cv,
    const float* __restrict__ kb, const float* __restrict__ vb,
    const float* __restrict__ mask,
    float* __restrict__ attnT) {
  __shared__ float probs[8][SEQL];  // 128 KB of the 320 KB LDS
  __shared__ float qs[8][HDIM];
  __shared__ float red[9];
  __shared__ float invd[8];
  __shared__ float maxsh;
  const int b = blockIdx.x >> 1;
  const int kvh = blockIdx.x & 1;
  const int tid = threadIdx.x;

  for (int i = tid; i < 8 * HDIM; i += 256) {
    int hh = i / HDIM, d = i % HDIM;
    qs[hh][d] = q[(size_t)b * (NHEADS * HDIM) + (kvh * 8 + hh) * HDIM + d];
  }
  __syncthreads();

  const float scale = 0.08838834764831845f;  // 1/sqrt(128)
  for (int p = tid; p < SEQL; p += 256) {
    const float4* krow;
    if (p < SEQL - 1)
      krow = (const float4*)(ck + (((size_t)b * SEQL + (p + 1)) * NKV + kvh) * HDIM);
    else
      krow = (const float4*)(kb + (size_t)b * (NKV * HDIM) + kvh * HDIM);
    float acc[8] = {0.f, 0.f, 0.f, 0.f, 0.f, 0.f, 0.f, 0.f};
#pragma unroll 4
    for (int c = 0; c < HDIM / 4; ++c) {
      float4 k4 = krow[c];
#pragma unroll
      for (int hh = 0; hh < 8; ++hh)
        acc[hh] += k4.x * qs[hh][4 * c] + k4.y * qs[hh][4 * c + 1] +
                   k4.z * qs[hh][4 * c + 2] + k4.w * qs[hh][4 * c + 3];
    }
    const float mk = mask[p];
#pragma unroll
    for (int hh = 0; hh < 8; ++hh) probs[hh][p] = acc[hh] * scale + mk;
  }
  __syncthreads();

  for (int hh = 0; hh < 8; ++hh) {
    float mx = -INFINITY;
    for (int p = tid; p < SEQL; p += 256) mx = fmaxf(mx, probs[hh][p]);
    for (int off = 16; off > 0; off >>= 1) mx = fmaxf(mx, __shfl_xor(mx, off, 32));
    if ((tid & 31) == 0) red[tid >> 5] = mx;
    __syncthreads();
    if (tid == 0) {
      float t = red[0];
      for (int i = 1; i < 8; ++i) t = fmaxf(t, red[i]);
      maxsh = t;
    }
    __syncthreads();
    const float gm = maxsh;
    float s = 0.f;
    for (int p = tid; p < SEQL; p += 256) {
      float e = expf(probs[hh][p] - gm);
      probs[hh][p] = e;
      s += e;
    }
    for (int off = 16; off > 0; off >>= 1) s += __shfl_xor(s, off, 32);
    if ((tid & 31) == 0) red[tid >> 5] = s;
    __syncthreads();
    if (tid == 0) {
      float t = 0.f;
      for (int i = 0; i < 8; ++i) t += red[i];
      invd[hh] = 1.f / t;
    }
    __syncthreads();
  }

  const int d = tid & (HDIM - 1);
  const int g = tid >> 7;
  float acc[8] = {0.f, 0.f, 0.f, 0.f, 0.f, 0.f, 0.f, 0.f};
  for (int p = g; p < SEQL; p += 2) {
    float vv;
    if (p < SEQL - 1)
      vv = cv[(((size_t)b * SEQL + (p + 1)) * NKV + kvh) * HDIM + d];
    else
      vv = vb[(size_t)b * (NKV * HDIM) + kvh * HDIM + d];
#pragma unroll
    for (int hh = 0; hh < 8; ++hh) acc[hh] += probs[hh][p] * vv;
  }
  __syncthreads();
  if (g == 1)
    for (int hh = 0; hh < 8; ++hh) qs[hh][d] = acc[hh];
  __syncthreads();
  if (g == 0)
    for (int hh = 0; hh < 8; ++hh) {
      const int e = (kvh * 8 + hh) * HDIM + d;
      attnT[apack(e, b)] = (acc[hh] + qs[hh][d]) * invd[hh];
    }
}

// ---------------------------------------------------------------------------
// silu(gate) * up, written as pair-interleaved transposed panel
// ---------------------------------------------------------------------------
__global__ __launch_bounds__(256) void silumulT_kernel(
    const float* __restrict__ gate, const float* __restrict__ up,
    float* __restrict__ guT) {
  int idx = blockIdx.x * blockDim.x + threadIdx.x;
  if (idx >= BATCH * IDIM) return;
  int b = idx / IDIM, i = idx % IDIM;
  float gv = gate[idx];
  float s = gv / (1.f + expf(-gv));
  guT[apack(i, b)] = s * up[idx];
}

// ---------------------------------------------------------------------------
extern "C" void kernel_launch(void* const* d_in, const int* in_sizes, int n_in,
                              void* d_out, int out_size, void* d_ws, size_t ws_size,
                              hipStream_t stream) {
  const float* x    = (const float*)d_in[0];
  const float* mask = (const float*)d_in[1];
  const float* ck   = (const float*)d_in[2];
  const float* cv   = (const float*)d_in[3];
  const float* fc   = (const float*)d_in[4];
  const float* wq   = (const float*)d_in[5];
  const float* bq   = (const float*)d_in[6];
  const float* wk   = (const float*)d_in[7];
  const float* bk   = (const float*)d_in[8];
  const float* wv   = (const float*)d_in[9];
  const float* bv   = (const float*)d_in[10];
  const float* wo   = (const float*)d_in[11];
  const float* wg   = (const float*)d_in[12];
  const float* wu   = (const float*)d_in[13];
  const float* wd   = (const float*)d_in[14];
  const float* ln1  = (const float*)d_in[15];
  const float* ln2  = (const float*)d_in[16];

  float* out = (float*)d_out;                           // [32][2048]
  float* nk  = out + (size_t)BATCH * DMODEL;            // new_cache_k
  float* nv  = nk + (size_t)BATCH * SEQL * NKV * HDIM;  // new_cache_v

  float* ws = (float*)d_ws;  // ~5.6 MB fp32 scratch
  float* xnT   = ws; ws += (size_t)DMODEL * BATCH;
  float* qbuf  = ws; ws += (size_t)BATCH * NHEADS * HDIM;
  float* kbuf  = ws; ws += (size_t)BATCH * NKV * HDIM;
  float* vbuf  = ws; ws += (size_t)BATCH * NKV * HDIM;
  float* attnT = ws; ws += (size_t)NHEADS * HDIM * BATCH;
  float* hbuf  = ws; ws += (size_t)BATCH * DMODEL;
  float* hnT   = ws; ws += (size_t)DMODEL * BATCH;
  float* gbuf  = ws; ws += (size_t)BATCH * IDIM;
  float* ubuf  = ws; ws += (size_t)BATCH * IDIM;
  float* guT   = ws; ws += (size_t)IDIM * BATCH;

  rmsnorm_T_kernel<<<BATCH, 256, 0, stream>>>(x, ln1, xnT);

  gemm_wmma_kernel<<<NHEADS * HDIM / 16 / 8, 256, 0, stream>>>(
      xnT, wq, bq, nullptr, qbuf, DMODEL, NHEADS * HDIM);
  gemm_wmma_kernel<<<NKV * HDIM / 16 / 8, 256, 0, stream>>>(
      xnT, wk, bk, nullptr, kbuf, DMODEL, NKV * HDIM);
  gemm_wmma_kernel<<<NKV * HDIM / 16 / 8, 256, 0, stream>>>(
      xnT, wv, bv, nullptr, vbuf, DMODEL, NKV * HDIM);

  rope_kernel<<<(BATCH * (NHEADS + NKV) * (HDIM / 2) + 255) / 256, 256, 0, stream>>>(
      qbuf, kbuf, fc);

  {
    size_t n4 = 2ull * BATCH * SEQL * NKV * HDIM / 4;
    cache_shift_kernel<<<(unsigned)((n4 + 255) / 256), 256, 0, stream>>>(
        (const float4*)ck, (const float4*)cv, (const float4*)kbuf,
        (const float4*)vbuf, (float4*)nk, (float4*)nv);
  }

  attn_kernel<<<BATCH * NKV, 256, 0, stream>>>(qbuf, ck, cv, kbuf, vbuf, mask, attnT);

  gemm_wmma_kernel<<<DMODEL / 16 / 8, 256, 0, stream>>>(
      attnT, wo, nullptr, x, hbuf, DMODEL, DMODEL);

  rmsnorm_T_kernel<<<BATCH, 256, 0, stream>>>(hbuf, ln2, hnT);

  gemm_wmma_kernel<<<IDIM / 16 / 8, 256, 0, stream>>>(
      hnT, wg, nullptr, nullptr, gbuf, DMODEL, IDIM);
  gemm_wmma_kernel<<<IDIM / 16 / 8, 256, 0, stream>>>(
      hnT, wu, nullptr, nullptr, ubuf, DMODEL, IDIM);

  silumulT_kernel<<<(BATCH * IDIM + 255) / 256, 256, 0, stream>>>(gbuf, ubuf, guT);

  gemm_wmma_kernel<<<DMODEL / 16 / 8, 256, 0, stream>>>(
      guT, wd, nullptr, hbuf, out, IDIM, DMODEL);
}